// PathDevelopmentNetwork_35227321761795
// MI455X (gfx1250) — compile-verified
//
#include <hip/hip_runtime.h>
#include <cstdint>
#include <cstddef>

#define T 2048
#define NFULL 6142      // 3T-2 = 2*37*83
#define BS 64
#define NC 576          // BS*9
#define NC2 1152        // 2*NC  (Xr | Xir columns)
#define NC4 2304        // 4*NC  (AA|AB|BA|BB columns)
#define TWO_PI 6.28318530717958647692f

typedef __attribute__((ext_vector_type(2))) float v2f;
typedef __attribute__((ext_vector_type(8))) float v8f;

// ---------------------------------------------------------------- 3x3 helpers
__device__ __forceinline__ void mm3(const float* a, const float* b, float* c) {
#pragma unroll
  for (int i = 0; i < 3; ++i)
#pragma unroll
    for (int j = 0; j < 3; ++j)
      c[3*i+j] = a[3*i+0]*b[0+j] + a[3*i+1]*b[3+j] + a[3*i+2]*b[6+j];
}

__device__ __forceinline__ void cp3(const float* s, float* d) {
#pragma unroll
  for (int e = 0; e < 9; ++e) d[e] = s[e];
}

// expm of skew-symmetric 3x3 via Rodrigues (exact)
__device__ __forceinline__ void make_step(float* M, int t, int b,
                                          const float* __restrict__ x,
                                          const float* A1s, const float* A2s) {
  if (t == 0) {
#pragma unroll
    for (int e = 0; e < 9; ++e) M[e] = (e == 0 || e == 4 || e == 8) ? 1.f : 0.f;
    return;
  }
  const float z1 = x[((size_t)b * T + t) * 2 + 0];
  const float z2 = x[((size_t)b * T + t) * 2 + 1];
  float G[9];
#pragma unroll
  for (int e = 0; e < 9; ++e) G[e] = A1s[e] * z1 + A2s[e] * z2;
  const float w1 = G[7], w2 = G[2], w3 = G[3];
  const float th2 = w1*w1 + w2*w2 + w3*w3;
  float s, c2;
  if (th2 < 1e-12f) {
    s  = 1.f  - th2 * (1.f / 6.f);
    c2 = 0.5f - th2 * (1.f / 24.f);
  } else {
    const float th = sqrtf(th2);
    s  = sinf(th) / th;
    c2 = (1.f - cosf(th)) / th2;
  }
  float G2[9];
  mm3(G, G, G2);
#pragma unroll
  for (int e = 0; e < 9; ++e)
    M[e] = ((e == 0 || e == 4 || e == 8) ? 1.f : 0.f) + s * G[e] + c2 * G2[e];
}

// ---------------------------------------------------------------- tiny FNNs
__global__ __launch_bounds__(256) void fnn_kernel(
    const float* w1f, const float* b1f, const float* w2f, const float* b2f,
    const float* w1g, const float* b1g, const float* w2g, const float* b2g,
    const float* w1p, const float* b1p, const float* w2p, const float* b2p,
    const float* w1q, const float* b1q, const float* w2q, const float* b2q,
    float* __restrict__ h) {
  const int idx = blockIdx.x * 256 + threadIdx.x;
  if (idx >= 4 * T) return;
  const int j = idx >> 11, i = idx & (T - 1);
  const float *w1, *b1, *w2, *b2;
  float x;
  switch (j) {
    case 0:  w1 = w1f; b1 = b1f; w2 = w2f; b2 = b2f; x = (float)(T - 1 - i); break;
    case 1:  w1 = w1g; b1 = b1g; w2 = w2g; b2 = b2g; x = (float)i;           break;
    case 2:  w1 = w1p; b1 = b1p; w2 = w2p; b2 = b2p; x = (float)(T - 1 - i); break;
    default: w1 = w1q; b1 = b1q; w2 = w2q; b2 = b2q; x = (float)i;           break;
  }
  float z = b2[0];
#pragma unroll
  for (int q = 0; q < 5; ++q) z += w2[q] * tanhf(w1[q] * x + b1[q]);
  h[idx] = z * expf(-5.0f * x);
}

// DFT of the (time-padded) filters evaluated only at the shifted bins m=k+T-1.
// Exact integer phase reduction: r = (m*(tau+T-1)) mod NFULL, incremental.
__global__ __launch_bounds__(256) void dft_h_kernel(const float* __restrict__ h,
                                                    float* __restrict__ Hre,
                                                    float* __restrict__ Him) {
  const int idx = blockIdx.x * 256 + threadIdx.x;
  if (idx >= 4 * T) return;
  const int j = idx >> 11, k = idx & (T - 1);
  const float* hv = h + (size_t)j * T;
  const unsigned m = (unsigned)k + (T - 1);
  const unsigned step = m % NFULL;
  unsigned r = (unsigned)(((unsigned long long)m * (T - 1)) % NFULL);
  float re = 0.f, im = 0.f;
  for (int tau = 0; tau < T; ++tau) {
    const float ang = (float)r * (TWO_PI / (float)NFULL);
    float sn, cs;
    __sincosf(ang, &sn, &cs);
    const float hv_t = hv[tau];
    re += hv_t * cs;
    im -= hv_t * sn;          // e^{-i ang}
    r += step;
    if (r >= NFULL) r -= NFULL;
  }
  Hre[idx] = re;
  Him[idx] = im;
}

// Stage-1 DFT matrix: D1c = cos(2*pi*k*u/T), D1sn = -sin(...)  (so Yim = D1sn*x)
__global__ __launch_bounds__(256) void twiddle1_kernel(float* __restrict__ D1c,
                                                       float* __restrict__ D1sn) {
  const int idx = blockIdx.x * 256 + threadIdx.x;
  const int k = idx >> 11, u = idx & (T - 1);
  const unsigned r = ((unsigned)k * (unsigned)u) & (T - 1);
  const float ang = (float)r * (TWO_PI / (float)T);
  float sn, cs;
  __sincosf(ang, &sn, &cs);
  D1c[idx] = cs;
  D1sn[idx] = -sn;
}

// Stage-2 restricted-ifft operator: W[n'][k] = e^{+i*2*pi*((k+T-1)*(n'+2T-2) mod N)/N}/N
__global__ __launch_bounds__(256) void twiddle2_kernel(float* __restrict__ W2c,
                                                       float* __restrict__ W2s) {
  const int idx = blockIdx.x * 256 + threadIdx.x;
  const int np = idx >> 11, k = idx & (T - 1);
  const unsigned m = (unsigned)k + (T - 1);
  const unsigned n = (unsigned)np + 2 * T - 2;
  const unsigned r = (m * n) % NFULL;          // < 2^25, exact in u32
  const float ang = (float)r * (TWO_PI / (float)NFULL);
  float sn, cs;
  __sincosf(ang, &sn, &cs);
  W2c[idx] = cs * (1.0f / (float)NFULL);
  W2s[idx] = sn * (1.0f / (float)NFULL);
}

// ---------------------------------------------------------------- PathDev scan
// One block per batch; 256 threads x 8 steps; Hillis-Steele prefix product of
// 3x3 matrices in LDS.  Writes Xr into cols [0,576) and Xir = Xr^T (orthogonal
// inverse) into cols [576,1152) of the stage-1 GEMM RHS.
__global__ __launch_bounds__(256) void pathdev_kernel(const float* __restrict__ x_data,
                                                      const float* __restrict__ A1,
                                                      const float* __restrict__ A2,
                                                      float* __restrict__ xcols) {
  __shared__ float S[2][256][9];
  const int b = blockIdx.x, tid = threadIdx.x;

  float A1s[9], A2s[9];
#pragma unroll
  for (int i = 0; i < 3; ++i)
#pragma unroll
    for (int j = 0; j < 3; ++j) {
      A1s[3*i+j] = A1[3*i+j] - A1[3*j+i];
      A2s[3*i+j] = A2[3*i+j] - A2[3*j+i];
    }

  // pass 1: local product of 8 steps
  float P[9] = {1,0,0, 0,1,0, 0,0,1};
#pragma unroll 2
  for (int jj = 0; jj < 8; ++jj) {
    const int t = tid * 8 + jj;
    float M[9], tmp[9];
    make_step(M, t, b, x_data, A1s, A2s);
    mm3(P, M, tmp);
    cp3(tmp, P);
  }
#pragma unroll
  for (int e = 0; e < 9; ++e) S[0][tid][e] = P[e];

  // inclusive scan (matrix product, time order = left to right)
  int src = 0;
  for (int d = 1; d < 256; d <<= 1) {
    __syncthreads();
    float R[9];
    if (tid >= d) mm3(S[src][tid - d], S[src][tid], R);
    else          cp3(S[src][tid], R);
#pragma unroll
    for (int e = 0; e < 9; ++e) S[src ^ 1][tid][e] = R[e];
    src ^= 1;
  }
  __syncthreads();

  float pref[9];
  if (tid == 0) {
#pragma unroll
    for (int e = 0; e < 9; ++e) pref[e] = (e == 0 || e == 4 || e == 8) ? 1.f : 0.f;
  } else {
    cp3(S[src][tid - 1], pref);
  }

  // pass 2: recompute local steps, emit prefixes
#pragma unroll 2
  for (int jj = 0; jj < 8; ++jj) {
    const int t = tid * 8 + jj;
    float M[9], tmp[9];
    make_step(M, t, b, x_data, A1s, A2s);
    mm3(pref, M, tmp);
    cp3(tmp, pref);
    const size_t base = (size_t)t * NC2 + (size_t)b * 9;
#pragma unroll
    for (int e = 0; e < 9; ++e) {
      xcols[base + e] = pref[e];                                 // Xr
      xcols[base + NC + e] = pref[3 * (e % 3) + (e / 3)];        // Xir = Xr^T
    }
  }
}

// ---------------------------------------------------------------- per-bin H scale
__global__ __launch_bounds__(256) void scale_kernel(const float* __restrict__ Yre,
                                                    const float* __restrict__ Yim,
                                                    const float* __restrict__ Hre,
                                                    const float* __restrict__ Him,
                                                    float* __restrict__ Rre,
                                                    float* __restrict__ Rim) {
  const size_t idx = (size_t)blockIdx.x * 256 + threadIdx.x;
  const int k = (int)(idx / NC4);
  const int c = (int)(idx % NC4);
  const int g = c / NC;               // 0:AA 1:AB 2:BA 3:BB
  const int cc = c - g * NC;
  const int ycol = (g == 0 || g == 2) ? NC + cc : cc;   // Xir for AA/BA, Xr for AB/BB
  const float yre = Yre[(size_t)k * NC2 + ycol];
  const float yim = Yim[(size_t)k * NC2 + ycol];
  const float hre = Hre[g * T + k];   // filter order f,g,fp,gp == group order
  const float him = Him[g * T + k];
  Rre[idx] = yre * hre - yim * him;
  Rim[idx] = yre * him + yim * hre;
}

// ---------------------------------------------------------------- WMMA GEMM
// CPLX=false: C0 = A0*B0, C1 = A1*B0            (two real outputs, shared B)
// CPLX=true : C0 = A0*B0 - A1*B1, C1 = A0*B1 + A1*B0   (complex)
// Block tile 64(M) x 128(N); 8 waves, 32x32 per wave; K pipelined 16 at a time
// through double-buffered LDS (global float4 loads for tile i+1 overlap the
// WMMA burst on tile i; one barrier per K-tile).
template <bool CPLX>
__global__ __launch_bounds__(256) void gemm_wmma_kernel(
    const float* __restrict__ A0, const float* __restrict__ A1,
    const float* __restrict__ B0, const float* __restrict__ B1,
    float* __restrict__ C0, float* __restrict__ C1,
    int K, int ldb) {
  // A stride 20 floats: rows 16B-aligned for b128 stores; column reads hit 16
  // distinct banks (gcd(20,64)=4 -> period-16 cycle).
  __shared__ float lA0[2][64][20];
  __shared__ float lA1[2][64][20];
  __shared__ float lB0[2][16][128];
  __shared__ float lB1[2][16][128];

  const int tid  = threadIdx.x;
  const int lane = tid & 31;
  const int lo   = lane & 15;
  const int hi   = lane >> 4;
  const int w    = tid >> 5;
  const int waveM = w >> 2;         // 0..1
  const int waveN = w & 3;          // 0..3
  const int mBlock = blockIdx.x * 64;
  const int nBlock = blockIdx.y * 128;

  // staging geometry: A tile 64x16 = 256 float4 (1/thread); B tile 16x128 =
  // 512 float4 (2/thread)
  const int rA = tid >> 2,        cA = (tid & 3) * 4;
  const int rB0 = tid >> 5,       cB0 = (tid & 31) * 4;
  const int rB1 = rB0 + 8;        // (tid + 256) >> 5
  const int cB1 = cB0;

  float4 ra0, ra1, rb00, rb01, rb10, rb11;

  auto load_tile = [&](int k0) {
    ra0  = *(const float4*)&A0[(size_t)(mBlock + rA) * K + k0 + cA];
    ra1  = *(const float4*)&A1[(size_t)(mBlock + rA) * K + k0 + cA];
    rb00 = *(const float4*)&B0[(size_t)(k0 + rB0) * ldb + nBlock + cB0];
    rb01 = *(const float4*)&B0[(size_t)(k0 + rB1) * ldb + nBlock + cB1];
    if (CPLX) {
      rb10 = *(const float4*)&B1[(size_t)(k0 + rB0) * ldb + nBlock + cB0];
      rb11 = *(const float4*)&B1[(size_t)(k0 + rB1) * ldb + nBlock + cB1];
    }
  };
  auto store_tile = [&](int buf) {
    *(float4*)&lA0[buf][rA][cA] = ra0;
    *(float4*)&lA1[buf][rA][cA] = ra1;
    *(float4*)&lB0[buf][rB0][cB0] = rb00;
    *(float4*)&lB0[buf][rB1][cB1] = rb01;
    if (CPLX) {
      *(float4*)&lB1[buf][rB0][cB0] = rb10;
      *(float4*)&lB1[buf][rB1][cB1] = rb11;
    }
  };

  v8f acc[2][2][2];
#pragma unroll
  for (int mt = 0; mt < 2; ++mt)
#pragma unroll
    for (int nt = 0; nt < 2; ++nt)
#pragma unroll
      for (int p = 0; p < 2; ++p) acc[mt][nt][p] = (v8f)0.f;

  const int ntiles = K / 16;
  load_tile(0);
  store_tile(0);

  for (int i = 0; i < ntiles; ++i) {
    __syncthreads();
    const int cur = i & 1;
    if (i + 1 < ntiles) load_tile((i + 1) * 16);
    if (i + 2 < ntiles) {   // warm L2 for the tile after next
      const int kp = (i + 2) * 16;
      __builtin_prefetch(&A0[(size_t)(mBlock + rA) * K + kp + cA], 0, 0);
      __builtin_prefetch(&B0[(size_t)(kp + rB0) * ldb + nBlock + cB0], 0, 0);
      if (CPLX) __builtin_prefetch(&B1[(size_t)(kp + rB0) * ldb + nBlock + cB0], 0, 0);
    }

#pragma unroll
    for (int kk = 0; kk < 16; kk += 4) {
      const int kc = kk + 2 * hi;
      v2f a0[2], a1[2], b0[2], b1v[2], b1n[2];
#pragma unroll
      for (int mt = 0; mt < 2; ++mt) {
        const int lm = waveM * 32 + mt * 16 + lo;
        a0[mt] = *(const v2f*)&lA0[cur][lm][kc];   // ds_load_b64 (8B aligned)
        a1[mt] = *(const v2f*)&lA1[cur][lm][kc];
      }
#pragma unroll
      for (int nt = 0; nt < 2; ++nt) {
        const int ln = waveN * 32 + nt * 16 + lo;
        b0[nt].x = lB0[cur][kc + 0][ln];
        b0[nt].y = lB0[cur][kc + 1][ln];
        if (CPLX) {
          b1v[nt].x = lB1[cur][kc + 0][ln];
          b1v[nt].y = lB1[cur][kc + 1][ln];
          b1n[nt] = -b1v[nt];   // fp32 WMMA has no operand negate -> 2 VALU ops
        }
      }
#pragma unroll
      for (int mt = 0; mt < 2; ++mt)
#pragma unroll
        for (int nt = 0; nt < 2; ++nt) {
          acc[mt][nt][0] = __builtin_amdgcn_wmma_f32_16x16x4_f32(
              false, a0[mt], false, b0[nt], (short)0, acc[mt][nt][0], false, false);
          acc[mt][nt][1] = __builtin_amdgcn_wmma_f32_16x16x4_f32(
              false, a1[mt], false, b0[nt], (short)0, acc[mt][nt][1], false, false);
          if (CPLX) {
            acc[mt][nt][0] = __builtin_amdgcn_wmma_f32_16x16x4_f32(
                false, a1[mt], false, b1n[nt], (short)0, acc[mt][nt][0], false, false);
            acc[mt][nt][1] = __builtin_amdgcn_wmma_f32_16x16x4_f32(
                false, a0[mt], false, b1v[nt], (short)0, acc[mt][nt][1], false, false);
          }
        }
    }

    if (i + 1 < ntiles) store_tile(cur ^ 1);
  }

#pragma unroll
  for (int mt = 0; mt < 2; ++mt)
#pragma unroll
    for (int nt = 0; nt < 2; ++nt)
#pragma unroll
      for (int v = 0; v < 8; ++v) {
        const int row = mBlock + waveM * 32 + mt * 16 + v + 8 * hi;
        const int col = nBlock + waveN * 32 + nt * 16 + lo;
        C0[(size_t)row * ldb + col] = acc[mt][nt][0][v];
        C1[(size_t)row * ldb + col] = acc[mt][nt][1][v];
      }
}

// ---------------------------------------------------------------- final combine
__global__ __launch_bounds__(256) void final_kernel(const float* __restrict__ Zre,
                                                    const float* __restrict__ Zim,
                                                    float* __restrict__ out) {
  const size_t idx = (size_t)blockIdx.x * 256 + threadIdx.x;
  const int b = (int)(idx / ((size_t)T * 9));
  const int rr = (int)(idx % ((size_t)T * 9));
  const int t = rr / 9, e = rr % 9;
  const int c = b * 9 + e;
  const size_t base = (size_t)t * NC4;
  const float aar = Zre[base + 0 * NC + c], aai = Zim[base + 0 * NC + c];
  const float abr = Zre[base + 1 * NC + c], abi = Zim[base + 1 * NC + c];
  const float bar = Zre[base + 2 * NC + c], bai = Zim[base + 2 * NC + c];
  const float bbr = Zre[base + 3 * NC + c], bbi = Zim[base + 3 * NC + c];
  out[idx] = aar * abr - aai * abi + bar * bbr - bai * bbi;
}

// ---------------------------------------------------------------- launcher
extern "C" void kernel_launch(void* const* d_in, const int* in_sizes, int n_in,
                              void* d_out, int out_size, void* d_ws, size_t ws_size,
                              hipStream_t stream) {
  (void)in_sizes; (void)n_in; (void)out_size; (void)ws_size;
  const float* x_data = (const float*)d_in[0];
  const float* A1 = (const float*)d_in[1];
  const float* A2 = (const float*)d_in[2];
  const float* W[16];
  for (int i = 0; i < 16; ++i) W[i] = (const float*)d_in[3 + i];
  float* out = (float*)d_out;

  float* p = (float*)d_ws;
  float* h    = p; p += 4 * T;
  float* Hre  = p; p += 4 * T;
  float* Him  = p; p += 4 * T;
  float* D1c  = p; p += (size_t)T * T;
  float* D1sn = p; p += (size_t)T * T;
  float* W2c  = p; p += (size_t)T * T;
  float* W2s  = p; p += (size_t)T * T;
  float* xcols= p; p += (size_t)T * NC2;
  float* Yre  = p; p += (size_t)T * NC2;
  float* Yim  = p; p += (size_t)T * NC2;
  float* Rre  = p; p += (size_t)T * NC4;
  float* Rim  = p; p += (size_t)T * NC4;
  float* Zre  = p; p += (size_t)T * NC4;
  float* Zim  = p; p += (size_t)T * NC4;

  fnn_kernel<<<(4 * T) / 256, 256, 0, stream>>>(
      W[0], W[1], W[2], W[3], W[4], W[5], W[6], W[7],
      W[8], W[9], W[10], W[11], W[12], W[13], W[14], W[15], h);
  dft_h_kernel<<<(4 * T) / 256, 256, 0, stream>>>(h, Hre, Him);
  twiddle1_kernel<<<((size_t)T * T) / 256, 256, 0, stream>>>(D1c, D1sn);
  twiddle2_kernel<<<((size_t)T * T) / 256, 256, 0, stream>>>(W2c, W2s);
  pathdev_kernel<<<BS, 256, 0, stream>>>(x_data, A1, A2, xcols);

  dim3 g1(T / 64, NC2 / 128);
  gemm_wmma_kernel<false><<<g1, 256, 0, stream>>>(D1c, D1sn, xcols, nullptr,
                                                  Yre, Yim, T, NC2);

  scale_kernel<<<((size_t)T * NC4) / 256, 256, 0, stream>>>(Yre, Yim, Hre, Him, Rre, Rim);

  dim3 g2(T / 64, NC4 / 128);
  gemm_wmma_kernel<true><<<g2, 256, 0, stream>>>(W2c, W2s, Rre, Rim,
                                                 Zre, Zim, T, NC4);

  final_kernel<<<((size_t)BS * T * 9) / 256, 256, 0, stream>>>(Zre, Zim, out);
}